// GC_SVDC_5617817223340
// MI455X (gfx1250) — compile-verified
//
#include <hip/hip_runtime.h>
#include <hip/hip_bf16.h>

#define QN    2048
#define BSZ   128
#define TMAXC 128
#define NB    256
#define NBLK  8
#define KSPL  4          // K-split of the step GEMM for occupancy
#define STEPF 0.01f

typedef __attribute__((ext_vector_type(2))) float v2f;
typedef __attribute__((ext_vector_type(8))) float v8f;

__device__ __forceinline__ v8f wmma4_f32(v2f a, v2f b, v8f c) {
  // V_WMMA_F32_16X16X4_F32 : D(16x16 f32) = A(16x4 f32) * B(4x16 f32) + C
  return __builtin_amdgcn_wmma_f32_16x16x4_f32(
      /*neg_a=*/false, a, /*neg_b=*/false, b,
      /*c_mod=*/(short)0, c, /*reuse_a=*/false, /*reuse_b=*/false);
}

// ---------------- Wm = W with diagonal 256x256 blocks zeroed ----------------
__global__ void k_mask_w(const float* __restrict__ W, float* __restrict__ Wm) {
  size_t i = (size_t)blockIdx.x * blockDim.x + threadIdx.x;   // < QN*QN
  int r = (int)(i >> 11);
  int c = (int)(i & (QN - 1));
  float v = W[i];
  Wm[i] = ((r >> 8) == (c >> 8)) ? 0.0f : v;
}

// ---------------- Invert the 8 diagonal 256x256 blocks of P in LDS ----------
// One workgroup per block; whole 256x256 tile (256KB) + pivot column in LDS
// (fits the 320KB WGP LDS). In-place Gauss-Jordan, no pivoting.
__global__ void k_block_inv(const float* __restrict__ P, float* __restrict__ Pinv) {
  extern __shared__ float smem[];          // NB*NB tile + NB pivot column
  float* sA   = smem;
  float* colk = smem + NB * NB;
  const int bi = blockIdx.x;
  const float* Pb = P + (size_t)bi * NB * QN + (size_t)bi * NB;

  for (int i = threadIdx.x; i < NB * NB; i += blockDim.x) {
    int r = i >> 8, c = i & 255;
    sA[i] = Pb[(size_t)r * QN + c];
  }
  __syncthreads();

  for (int k = 0; k < NB; ++k) {
    const float pivinv = 1.0f / sA[k * NB + k];
    __syncthreads();
    for (int j = threadIdx.x; j < NB; j += blockDim.x) {
      float v = sA[k * NB + j] * pivinv;
      if (j == k) v = pivinv;
      sA[k * NB + j] = v;
    }
    __syncthreads();
    for (int r = threadIdx.x; r < NB; r += blockDim.x) colk[r] = sA[r * NB + k];
    __syncthreads();
    for (int i = threadIdx.x; i < NB * NB; i += blockDim.x) {
      int r = i >> 8, c = i & 255;
      if (r != k) {
        float dum = colk[r];
        float v = sA[i] - dum * sA[k * NB + c];
        if (c == k) v = -dum * pivinv;
        sA[i] = v;
      }
    }
    __syncthreads();
  }

  float* Po = Pinv + (size_t)bi * NB * NB;
  for (int i = threadIdx.x; i < NB * NB; i += blockDim.x) Po[i] = sA[i];
}

// ---------------- T = Wm^T @ P  (2048x2048x2048 fp32 WMMA GEMM) -------------
// Wave tile: 16(M) x 64(N), K step 4.
__global__ void k_gemm_T(const float* __restrict__ Wm, const float* __restrict__ P,
                         float* __restrict__ T) {
  const int lane  = threadIdx.x & 31;
  const int gw    = blockIdx.x * (blockDim.x >> 5) + (threadIdx.x >> 5);
  const int NT_N  = QN / 64;
  const int mbase = (gw / NT_N) * 16;
  const int nbase = (gw % NT_N) * 64;
  const int row   = mbase + (lane & 15);
  const int khalf = (lane >> 4) * 2;
  const int c     = lane & 15;

  v8f a0 = {}, a1 = {}, a2 = {}, a3 = {};
  for (int k = 0; k < QN; k += 4) {
    const int kk = k + khalf;
    v2f af;                                        // A = Wm^T : A[row][kk] = Wm[kk][row]
    af.x = Wm[(size_t)kk * QN + row];
    af.y = Wm[(size_t)(kk + 1) * QN + row];
    const float* Bp = P + (size_t)kk * QN + nbase + c;
    v2f b0, b1, b2, b3;
    b0.x = Bp[0];   b0.y = Bp[QN + 0];
    b1.x = Bp[16];  b1.y = Bp[QN + 16];
    b2.x = Bp[32];  b2.y = Bp[QN + 32];
    b3.x = Bp[48];  b3.y = Bp[QN + 48];
    a0 = wmma4_f32(af, b0, a0);
    a1 = wmma4_f32(af, b1, a1);
    a2 = wmma4_f32(af, b2, a2);
    a3 = wmma4_f32(af, b3, a3);
  }
  const int col = nbase + c;
  const int rb  = mbase + ((lane >> 4) << 3);
#pragma unroll
  for (int r = 0; r < 8; ++r) {
    size_t i0 = (size_t)(rb + r) * QN + col;
    T[i0 +  0] = a0[r];
    T[i0 + 16] = a1[r];
    T[i0 + 32] = a2[r];
    T[i0 + 48] = a3[r];
  }
}

// ------- M = A + Wm - Pinv_blk @ T   (block-diag solve, K=256 per row-block) -
__global__ void k_gemm_M(const float* __restrict__ Pinv, const float* __restrict__ T,
                         const float* __restrict__ A, const float* __restrict__ Wm,
                         float* __restrict__ Mm) {
  const int lane  = threadIdx.x & 31;
  const int gw    = blockIdx.x * (blockDim.x >> 5) + (threadIdx.x >> 5);
  const int NT_N  = QN / 64;
  const int mbase = (gw / NT_N) * 16;
  const int nbase = (gw % NT_N) * 64;
  const int row   = mbase + (lane & 15);
  const int khalf = (lane >> 4) * 2;
  const int c     = lane & 15;
  const int bi    = mbase >> 8;
  const int ri    = row & 255;
  const float* Pb = Pinv + (size_t)bi * NB * NB;

  v8f a0 = {}, a1 = {}, a2 = {}, a3 = {};
  for (int k = 0; k < NB; k += 4) {
    v2f af = *(const v2f*)(Pb + (size_t)ri * NB + k + khalf);
    const float* Bp = T + (size_t)(bi * NB + k + khalf) * QN + nbase + c;
    v2f b0, b1, b2, b3;
    b0.x = Bp[0];   b0.y = Bp[QN + 0];
    b1.x = Bp[16];  b1.y = Bp[QN + 16];
    b2.x = Bp[32];  b2.y = Bp[QN + 32];
    b3.x = Bp[48];  b3.y = Bp[QN + 48];
    a0 = wmma4_f32(af, b0, a0);
    a1 = wmma4_f32(af, b1, a1);
    a2 = wmma4_f32(af, b2, a2);
    a3 = wmma4_f32(af, b3, a3);
  }
  const int col = nbase + c;
  const int rb  = mbase + ((lane >> 4) << 3);
#pragma unroll
  for (int r = 0; r < 8; ++r) {
    size_t i0 = (size_t)(rb + r) * QN + col;
    Mm[i0 +  0] = A[i0 +  0] + Wm[i0 +  0] - a0[r];
    Mm[i0 + 16] = A[i0 + 16] + Wm[i0 + 16] - a1[r];
    Mm[i0 + 32] = A[i0 + 32] + Wm[i0 + 32] - a2[r];
    Mm[i0 + 48] = A[i0 + 48] + Wm[i0 + 48] - a3[r];
  }
}

// ------------- init: Xcur = X0^T, and write t=0 slice of the output ---------
__global__ void k_init_x(const float* __restrict__ X0, float* __restrict__ Xa,
                         float* __restrict__ out) {
  size_t tid = (size_t)blockIdx.x * blockDim.x + threadIdx.x;  // < QN*BSZ
  int b = (int)(tid >> 11);
  int f = (int)(tid & (QN - 1));
  float v = X0[(size_t)b * QN + f];
  out[((size_t)b * TMAXC) * QN + f] = v;        // out[b][0][f]
  Xa[(size_t)f * BSZ + b] = v;                  // Xi0 = X0^T (QN x BS)
}

// ---- step phase 1: partial[ks] = M[:,ks] @ Xi[ks] + B[:,ks] @ relu(Xi[ks]) -
// Fused M/B passes; X fragments loaded once and shared by both row strips.
// Wave tile: 32(feature) x 32(batch), K-range = 512 per split.
// Per K=4 step: 4 x b64 A-loads + 4 x b32 X-loads -> 8 WMMAs (1:1 load:wmma).
// Waves: (2048/32) * (128/32) * KSPL = 1024 -> 128 workgroups of 8 waves.
__global__ void k_step_part(const float* __restrict__ Mm, const float* __restrict__ Bm,
                            const float* __restrict__ Xi, float* __restrict__ part) {
  const int lane  = threadIdx.x & 31;
  const int gw    = blockIdx.x * (blockDim.x >> 5) + (threadIdx.x >> 5);
  const int mbase = (gw >> 4) * 32;            // 64 row strips of 32
  const int ct    = (gw >> 2) & 3;             // 4 col tiles
  const int ks    = gw & 3;                    // 4 K splits
  const int nbase = ct * 32;
  const int kbase = ks * (QN / KSPL);
  const int row0  = mbase + (lane & 15);
  const int row1  = row0 + 16;
  const int khalf = (lane >> 4) * 2;
  const int c     = lane & 15;

  v8f acc00 = {}, acc01 = {}, acc10 = {}, acc11 = {};
  for (int k = kbase; k < kbase + QN / KSPL; k += 4) {
    v2f am0 = *(const v2f*)(Mm + (size_t)row0 * QN + k + khalf);
    v2f am1 = *(const v2f*)(Mm + (size_t)row1 * QN + k + khalf);
    v2f ab0 = *(const v2f*)(Bm + (size_t)row0 * QN + k + khalf);
    v2f ab1 = *(const v2f*)(Bm + (size_t)row1 * QN + k + khalf);
    const float* Xp = Xi + (size_t)(k + khalf) * BSZ + nbase + c;
    float x00 = Xp[0], x01 = Xp[BSZ + 0];
    float x10 = Xp[16], x11 = Xp[BSZ + 16];
    v2f b0; b0.x = x00; b0.y = x01;
    v2f b1; b1.x = x10; b1.y = x11;
    v2f r0; r0.x = fmaxf(x00, 0.0f); r0.y = fmaxf(x01, 0.0f);
    v2f r1; r1.x = fmaxf(x10, 0.0f); r1.y = fmaxf(x11, 0.0f);
    acc00 = wmma4_f32(am0, b0, acc00);
    acc01 = wmma4_f32(am0, b1, acc01);
    acc10 = wmma4_f32(am1, b0, acc10);
    acc11 = wmma4_f32(am1, b1, acc11);
    acc00 = wmma4_f32(ab0, r0, acc00);
    acc01 = wmma4_f32(ab0, r1, acc01);
    acc10 = wmma4_f32(ab1, r0, acc10);
    acc11 = wmma4_f32(ab1, r1, acc11);
  }
  float* Pp = part + (size_t)ks * QN * BSZ;
  const int col = nbase + c;
  const int rb0 = mbase + ((lane >> 4) << 3);
  const int rb1 = rb0 + 16;
#pragma unroll
  for (int r = 0; r < 8; ++r) {
    Pp[(size_t)(rb0 + r) * BSZ + col]      = acc00[r];
    Pp[(size_t)(rb0 + r) * BSZ + col + 16] = acc01[r];
    Pp[(size_t)(rb1 + r) * BSZ + col]      = acc10[r];
    Pp[(size_t)(rb1 + r) * BSZ + col + 16] = acc11[r];
  }
}

// ---- step phase 2: Xn = Xi + STEP*(sum_k part[k] + bx); write out[b][t][f] -
// Fixed-order summation -> deterministic across replays.
__global__ void k_step_reduce(const float* __restrict__ Xi, const float* __restrict__ part,
                              const float* __restrict__ bx, float* __restrict__ Xn,
                              float* __restrict__ out, int t) {
  size_t tid = (size_t)blockIdx.x * blockDim.x + threadIdx.x;  // < QN*BSZ
  int b = (int)(tid >> 11);
  int f = (int)(tid & (QN - 1));
  size_t xi = (size_t)f * BSZ + b;
  const size_t PS = (size_t)QN * BSZ;
  float s = part[xi] + part[PS + xi];
  s += part[2 * PS + xi] + part[3 * PS + xi];
  float v = Xi[xi] + STEPF * (s + bx[f]);
  Xn[xi] = v;
  out[((size_t)b * TMAXC + t) * QN + f] = v;    // f-contiguous HBM store
}

extern "C" void kernel_launch(void* const* d_in, const int* in_sizes, int n_in,
                              void* d_out, int out_size, void* d_ws, size_t ws_size,
                              hipStream_t stream) {
  (void)in_sizes; (void)n_in; (void)out_size; (void)ws_size;
  const float* X0 = (const float*)d_in[0];
  const float* A  = (const float*)d_in[1];
  const float* Bm = (const float*)d_in[2];
  const float* W  = (const float*)d_in[3];
  const float* P  = (const float*)d_in[4];
  const float* bx = (const float*)d_in[5];
  float* out = (float*)d_out;

  const size_t QQ = (size_t)QN * QN;
  const size_t XS = (size_t)QN * BSZ;
  float* ws   = (float*)d_ws;
  float* Wm   = ws;                                   // 16 MB
  float* T    = Wm + QQ;                              // 16 MB
  float* Mm   = T + QQ;                               // 16 MB
  float* Pinv = Mm + QQ;                              // 2 MB
  float* Xa   = Pinv + (size_t)NBLK * NB * NB;        // 1 MB
  float* Xb   = Xa + XS;                              // 1 MB
  float* Part = Xb + XS;                              // 4 MB (KSPL partials)

  const int lds_bytes = (NB * NB + NB) * (int)sizeof(float);   // 257 KB < 320 KB WGP LDS
  (void)hipFuncSetAttribute((const void*)k_block_inv,
                            hipFuncAttributeMaxDynamicSharedMemorySize, lds_bytes);

  // preprocessing
  k_mask_w<<<(int)(QQ / 256), 256, 0, stream>>>(W, Wm);
  k_block_inv<<<NBLK, 1024, lds_bytes, stream>>>(P, Pinv);
  {
    const int waves = (QN / 16) * (QN / 64);          // 4096 wave tiles
    k_gemm_T<<<waves / 8, 256, 0, stream>>>(Wm, P, T);
    k_gemm_M<<<waves / 8, 256, 0, stream>>>(Pinv, T, A, Wm, Mm);
  }
  // t = 0 slice + Xi0
  k_init_x<<<(QN * BSZ) / 256, 256, 0, stream>>>(X0, Xa, out);

  // 127 sequential Euler steps (ping-pong state, K-split partials)
  float* cur = Xa;
  float* nxt = Xb;
  const int part_waves = (QN / 32) * (BSZ / 32) * KSPL;   // 1024 waves
  for (int t = 1; t < TMAXC; ++t) {
    k_step_part<<<part_waves / 8, 256, 0, stream>>>(Mm, Bm, cur, Part);
    k_step_reduce<<<(QN * BSZ) / 256, 256, 0, stream>>>(cur, Part, bx, nxt, out, t);
    float* tmp = cur; cur = nxt; nxt = tmp;
  }
}